// Graph_Critic_Model_42159398978354
// MI455X (gfx1250) — compile-verified
//
#include <hip/hip_runtime.h>
#include <hip/hip_bf16.h>

typedef __attribute__((ext_vector_type(2))) float v2f;
typedef __attribute__((ext_vector_type(8))) float v8f;

#define NN   8192
#define HID  32
#define KSPLIT 8
#define KCHUNK 1024          // NN / KSPLIT
#define ROWS_PER_WG 128      // 8 waves * 16 rows
#define ROW_BLOCKS (NN / ROWS_PER_WG)   // 64
// LDS B-operand swizzle strides (floats): pad K-pair stride to 96 so the two
// lane-halves hit disjoint bank halves (96 % 64 == 32) -> conflict-free b64.
#define GSTR 192             // per K-group-of-4 stride
#define PSTR 96              // per K-pair stride

// ---------------------------------------------------------------------------
// Kernel 1: encoder  X = relu(relu([feat|act] @ We1 + be1) @ We2 + be2)
// ---------------------------------------------------------------------------
__global__ __launch_bounds__(256) void encoder_kernel(
    const float* __restrict__ F, const float* __restrict__ Act,
    const float* __restrict__ We1, const float* __restrict__ be1,
    const float* __restrict__ We2, const float* __restrict__ be2,
    float* __restrict__ X)
{
    const int i = blockIdx.x * 256 + threadIdx.x;
    float in[72];
#pragma unroll
    for (int j = 0; j < 64; ++j) in[j] = F[(size_t)i * 64 + j];
#pragma unroll
    for (int j = 0; j < 8; ++j)  in[64 + j] = Act[(size_t)i * 8 + j];

    float h1[HID];
    for (int h = 0; h < HID; ++h) {
        float acc = be1[h];
        for (int j = 0; j < 72; ++j) acc = fmaf(in[j], We1[j * HID + h], acc);
        h1[h] = fmaxf(acc, 0.0f);
    }
    for (int o = 0; o < HID; ++o) {
        float acc = be2[o];
#pragma unroll
        for (int h = 0; h < HID; ++h) acc = fmaf(h1[h], We2[h * HID + o], acc);
        X[(size_t)i * HID + o] = fmaxf(acc, 0.0f);
    }
}

// ---------------------------------------------------------------------------
// Kernel 2: deg rowsum of A (1st 268MB pass), dinv = rsqrt(deg+1),
// fused row-local S = (X @ Wg) * dinv.   One block per row.
// ---------------------------------------------------------------------------
__global__ __launch_bounds__(256) void deg_s_kernel(
    const float* __restrict__ A, const float* __restrict__ X,
    const float* __restrict__ Wg,
    float* __restrict__ dinv, float* __restrict__ S)
{
    __shared__ float red[256];
    const int row = blockIdx.x;
    const float4* Ar4 = (const float4*)(A + (size_t)row * NN);

    float s = 0.0f;
    for (int t = threadIdx.x; t < NN / 4; t += 256) {
        float4 v = Ar4[t];
        s += (v.x + v.y) + (v.z + v.w);
    }
    red[threadIdx.x] = s;
    __syncthreads();
    for (int off = 128; off > 0; off >>= 1) {
        if (threadIdx.x < off) red[threadIdx.x] += red[threadIdx.x + off];
        __syncthreads();
    }
    const float di = rsqrtf(red[0] + 1.0f);
    if (threadIdx.x == 0) dinv[row] = di;
    if (threadIdx.x < HID) {
        const int h = threadIdx.x;
        float acc = 0.0f;
#pragma unroll
        for (int j = 0; j < HID; ++j)
            acc = fmaf(X[(size_t)row * HID + j], Wg[j * HID + h], acc);
        S[(size_t)row * HID + h] = acc * di;   // bias bg applied post-aggregation
    }
}

// ---------------------------------------------------------------------------
// Kernel 3: Mpart[ks] = A[:, K slice] @ S[K slice, :]  via V_WMMA_F32_16X16X4_F32.
// Wave -> 16x32 tile (two v8f accumulators). grid = 64 row-blocks x 8 K-splits.
// S chunk is staged in LDS pre-swizzled into the WMMA B-operand layout:
//   element S[k][n]  ->  sS[(k>>2)*GSTR + ((k>>1)&1)*PSTR + n*2 + (k&1)]
// so each lane's B operand {B[k][n], B[k+1][n]} is one aligned ds_load_b64,
// bank-conflict-free, and needs no VALU shuffles in front of the WMMA.
// ---------------------------------------------------------------------------
__global__ __launch_bounds__(256) void spmm_wmma_kernel(
    const float* __restrict__ A, const float* __restrict__ S,
    float* __restrict__ Mpart)
{
    __shared__ float sS[16 * GSTR];   // 12 KB: 64 K-rows x 32 cols, swizzled

    const int rowBlock = blockIdx.x & (ROW_BLOCKS - 1);
    const int ksplit   = blockIdx.x >> 6;
    const int kBase    = ksplit * KCHUNK;

    const int wave = threadIdx.x >> 5;
    const int lane = threadIdx.x & 31;
    const int lm   = lane & 15;       // M index (A) / N index (B)
    const int lhi  = lane >> 4;       // selects K pair {0,1} vs {2,3}

    // cooperative-copy assignment: thread -> (K-group, K-pair, column quad)
    const int tg = threadIdx.x >> 4;          // 0..15
    const int tp = (threadIdx.x >> 3) & 1;    // 0..1
    const int tn = (threadIdx.x & 7) * 4;     // 0,4,...,28
    const int tk = tg * 4 + tp * 2;           // local k row (even)

    const int waveRow = rowBlock * ROWS_PER_WG + wave * 16;
    const float* Arow = A + (size_t)(waveRow + lm) * NN + kBase + 2 * lhi;

    v8f acc0 = {};   // N = 0..15
    v8f acc1 = {};   // N = 16..31

    for (int kc = 0; kc < KCHUNK; kc += 64) {
        __syncthreads();   // previous chunk's ds reads done
        {
            const float* s0 = S + (size_t)(kBase + kc + tk) * HID + tn;
            float4 r0 = *(const float4*)s0;          // row k,   cols tn..tn+3
            float4 r1 = *(const float4*)(s0 + HID);  // row k+1, cols tn..tn+3
            float* dst = sS + tg * GSTR + tp * PSTR + tn * 2;
            *(float4*)(dst)     = make_float4(r0.x, r1.x, r0.y, r1.y);
            *(float4*)(dst + 4) = make_float4(r0.z, r1.z, r0.w, r1.w);
        }
        __syncthreads();

#pragma unroll
        for (int j = 0; j < 16; ++j) {
            v2f a = *(const v2f*)(Arow + kc + j * 4); // 8B-aligned global_load_b64

            const float* bp = sS + j * GSTR + lhi * PSTR + lm * 2;
            v2f b0 = *(const v2f*)bp;          // {B[k][lm],    B[k+1][lm]}
            v2f b1 = *(const v2f*)(bp + 32);   // {B[k][16+lm], B[k+1][16+lm]}

            acc0 = __builtin_amdgcn_wmma_f32_16x16x4_f32(
                       false, a, false, b0, (short)0, acc0, false, false);
            acc1 = __builtin_amdgcn_wmma_f32_16x16x4_f32(
                       false, a, false, b1, (short)0, acc1, false, false);
        }
    }

    // C/D layout: VGPR r, lanes 0-15 -> M=r, lanes 16-31 -> M=r+8; N=lane&15
    float* Mp = Mpart + (size_t)ksplit * ((size_t)NN * HID);
#pragma unroll
    for (int r = 0; r < 8; ++r) {
        const int row = waveRow + r + 8 * lhi;
        Mp[(size_t)row * HID + lm]      = acc0[r];
        Mp[(size_t)row * HID + 16 + lm] = acc1[r];
    }
}

// ---------------------------------------------------------------------------
// Kernel 4: reduce K-partials + entire row-local tail:
// Xg = relu((M+S)*dinv + bg); G = relu(Xg@Wgd+bgd);
// P = relu([G,X]@Wp1+bp1); P = relu(P@Wp2+bp2); V = P@Wv+bv
// ---------------------------------------------------------------------------
__global__ __launch_bounds__(256) void tail_kernel(
    const float* __restrict__ Mpart, const float* __restrict__ S,
    const float* __restrict__ X, const float* __restrict__ dinv,
    const float* __restrict__ bg,
    const float* __restrict__ Wgd, const float* __restrict__ bgd,
    const float* __restrict__ Wp1, const float* __restrict__ bp1,
    const float* __restrict__ Wp2, const float* __restrict__ bp2,
    const float* __restrict__ Wv,  const float* __restrict__ bv,
    float* __restrict__ V)
{
    const int i = blockIdx.x * 256 + threadIdx.x;
    const float di = dinv[i];
    const size_t NH = (size_t)NN * HID;

    float a[HID], b[HID], x[HID];
#pragma unroll
    for (int h = 0; h < HID; ++h) x[h] = X[(size_t)i * HID + h];

#pragma unroll
    for (int h = 0; h < HID; ++h) {
        const size_t o = (size_t)i * HID + h;
        float m = 0.0f;
#pragma unroll
        for (int r = 0; r < KSPLIT; ++r) m += Mpart[(size_t)r * NH + o];
        a[h] = fmaxf(fmaf(m + S[o], di, bg[h]), 0.0f);     // Xg
    }
    for (int o = 0; o < HID; ++o) {                        // G = relu(Xg@Wgd+bgd)
        float acc = bgd[o];
#pragma unroll
        for (int h = 0; h < HID; ++h) acc = fmaf(a[h], Wgd[h * HID + o], acc);
        b[o] = fmaxf(acc, 0.0f);
    }
    for (int o = 0; o < HID; ++o) {                        // relu([G,X]@Wp1+bp1)
        float acc = bp1[o];
#pragma unroll
        for (int h = 0; h < HID; ++h) acc = fmaf(b[h], Wp1[h * HID + o], acc);
#pragma unroll
        for (int h = 0; h < HID; ++h) acc = fmaf(x[h], Wp1[(HID + h) * HID + o], acc);
        a[o] = fmaxf(acc, 0.0f);
    }
    for (int o = 0; o < HID; ++o) {                        // relu(P@Wp2+bp2)
        float acc = bp2[o];
#pragma unroll
        for (int h = 0; h < HID; ++h) acc = fmaf(a[h], Wp2[h * HID + o], acc);
        b[o] = fmaxf(acc, 0.0f);
    }
    float v = bv[0];
#pragma unroll
    for (int h = 0; h < HID; ++h) v = fmaf(b[h], Wv[h], v);
    V[i] = v;
}

// ---------------------------------------------------------------------------
extern "C" void kernel_launch(void* const* d_in, const int* in_sizes, int n_in,
                              void* d_out, int out_size, void* d_ws, size_t ws_size,
                              hipStream_t stream) {
    const float* features = (const float*)d_in[0];
    const float* adjacency = (const float*)d_in[1];
    /* d_in[2] mask: unused by reference */
    const float* action = (const float*)d_in[3];
    const float* We1 = (const float*)d_in[4];  const float* be1 = (const float*)d_in[5];
    const float* We2 = (const float*)d_in[6];  const float* be2 = (const float*)d_in[7];
    const float* Wg  = (const float*)d_in[8];  const float* bg  = (const float*)d_in[9];
    const float* Wgd = (const float*)d_in[10]; const float* bgd = (const float*)d_in[11];
    const float* Wp1 = (const float*)d_in[12]; const float* bp1 = (const float*)d_in[13];
    const float* Wp2 = (const float*)d_in[14]; const float* bp2 = (const float*)d_in[15];
    const float* Wv  = (const float*)d_in[16]; const float* bv  = (const float*)d_in[17];

    float* ws = (float*)d_ws;
    float* X     = ws;                       // 8192*32 = 262144 floats
    float* S     = ws + 262144;              // 262144
    float* dinv  = ws + 524288;              // 8192
    float* Mpart = ws + 532480;              // KSPLIT * 262144 (K-split partials)

    encoder_kernel<<<NN / 256, 256, 0, stream>>>(features, action, We1, be1, We2, be2, X);
    deg_s_kernel<<<NN, 256, 0, stream>>>(adjacency, X, Wg, dinv, S);
    spmm_wmma_kernel<<<ROW_BLOCKS * KSPLIT, 256, 0, stream>>>(adjacency, S, Mpart);
    tail_kernel<<<NN / 256, 256, 0, stream>>>(Mpart, S, X, dinv, bg, Wgd, bgd,
                                              Wp1, bp1, Wp2, bp2, Wv, bv, (float*)d_out);
}